// Encoder_78469052497925
// MI455X (gfx1250) — compile-verified
//
#include <hip/hip_runtime.h>

typedef __attribute__((ext_vector_type(2))) float v2f;
typedef __attribute__((ext_vector_type(8))) float v8f;

#define C_IN  256
#define C_HID 48
#define C_OUT 16

// ---------------- degree / normalization ----------------

__global__ void k_deg_init(float* __restrict__ deg, int n) {
  int i = blockIdx.x * blockDim.x + threadIdx.x;
  if (i < n) deg[i] = 1.0f;  // self loop
}

__global__ void k_deg_edges(const long long* __restrict__ dst, long long E,
                            float* __restrict__ deg) {
  long long i = (long long)blockIdx.x * blockDim.x + threadIdx.x;
  if (i < E) atomicAdd(&deg[(int)dst[i]], 1.0f);
}

__global__ void k_dinv(float* __restrict__ deg, int n) {
  int i = blockIdx.x * blockDim.x + threadIdx.x;
  if (i < n) deg[i] = rsqrtf(deg[i]);  // deg >= 1 always (self loops)
}

// ---------------- GEMM1: h1 = x @ W1  (fp32 WMMA 16x16x4) ----------------
// block = 256 threads = 8 waves; wave w owns row-tile (blockIdx.x*8+w), all 3 N-tiles.
__global__ void __launch_bounds__(256) k_gemm1(const float* __restrict__ x,
                                               const float* __restrict__ W1,
                                               float* __restrict__ h1, int M) {
  __shared__ float sW[C_IN * C_HID];  // 48 KB, L2->LDS once per block
  for (int i = threadIdx.x; i < C_IN * C_HID; i += 256) sW[i] = W1[i];
  __syncthreads();

  const int wave = threadIdx.x >> 5;
  const int lane = threadIdx.x & 31;
  const int rowTile = blockIdx.x * 8 + wave;
  const int row0 = rowTile << 4;
  if (row0 >= M) return;  // uniform per wave -> EXEC all-ones for WMMA

  const int mn = lane & 15;          // A: row index, B: col index
  const int kk = (lane >> 4) << 1;   // 0 or 2 (K sub-offset per ISA layout)
  const float* __restrict__ xr = x + (size_t)(row0 + mn) * C_IN;

  v8f acc0 = {}, acc1 = {}, acc2 = {};
  for (int k = 0; k < C_IN; k += 4) {
    v2f a;
    a.x = xr[k + kk];
    a.y = xr[k + kk + 1];
    const float* w0 = &sW[(size_t)(k + kk) * C_HID];
    const float* w1 = &sW[(size_t)(k + kk + 1) * C_HID];
    v2f b0; b0.x = w0[mn];      b0.y = w1[mn];
    v2f b1; b1.x = w0[16 + mn]; b1.y = w1[16 + mn];
    v2f b2; b2.x = w0[32 + mn]; b2.y = w1[32 + mn];
    acc0 = __builtin_amdgcn_wmma_f32_16x16x4_f32(false, a, false, b0, (short)0, acc0, false, false);
    acc1 = __builtin_amdgcn_wmma_f32_16x16x4_f32(false, a, false, b1, (short)0, acc1, false, false);
    acc2 = __builtin_amdgcn_wmma_f32_16x16x4_f32(false, a, false, b2, (short)0, acc2, false, false);
  }

  // D layout: VGPR r: lanes 0-15 -> M=r, N=lane; lanes 16-31 -> M=8+r, N=lane-16
  const int n = lane & 15;
  const int mbase = (lane >> 4) << 3;
  for (int r = 0; r < 8; ++r) {
    float* o = h1 + (size_t)(row0 + mbase + r) * C_HID + n;
    o[0]  = acc0[r];
    o[16] = acc1[r];
    o[32] = acc2[r];
  }
}

// ---------------- layer-1 aggregation ----------------

// init with self-loop contribution: agg1 = h1 * dinv^2 (non-atomic, also zero-fills)
__global__ void k_self1(const float* __restrict__ h1, const float* __restrict__ dinv,
                        float* __restrict__ agg1, int n) {
  int idx = blockIdx.x * blockDim.x + threadIdx.x;
  if (idx >= n * C_HID) return;
  int node = idx / C_HID;
  float di = dinv[node];
  agg1[idx] = h1[idx] * di * di;
}

// 16 lanes per edge, 3 features each -> coalesced 64B gathers & atomics
__global__ void k_scatter1(const long long* __restrict__ src,
                           const long long* __restrict__ dst, long long E,
                           const float* __restrict__ dinv,
                           const float* __restrict__ h1,
                           float* __restrict__ agg1) {
  long long t = (long long)blockIdx.x * blockDim.x + threadIdx.x;
  long long e = t >> 4;
  int f = (int)(t & 15);
  if (e >= E) return;
  if (f == 0 && e + 2048 < E) {
    __builtin_prefetch(&src[e + 2048], 0, 0);
    __builtin_prefetch(&dst[e + 2048], 0, 0);
  }
  int s = (int)src[e], d = (int)dst[e];
  float nrm = dinv[s] * dinv[d];
  const float* hs = h1 + (size_t)s * C_HID;
  float* od = agg1 + (size_t)d * C_HID;
  atomicAdd(&od[f],      hs[f]      * nrm);
  atomicAdd(&od[f + 16], hs[f + 16] * nrm);
  atomicAdd(&od[f + 32], hs[f + 32] * nrm);
}

// ---------------- GEMM2: h2 = relu(agg1 + b1) @ W2  (fused epilogue-of-layer-1) ----
__global__ void __launch_bounds__(256) k_gemm2(const float* __restrict__ agg1,
                                               const float* __restrict__ b1,
                                               const float* __restrict__ W2,
                                               float* __restrict__ h2, int M) {
  __shared__ float sW[C_HID * C_OUT];
  __shared__ float sB[C_HID];
  for (int i = threadIdx.x; i < C_HID * C_OUT; i += 256) sW[i] = W2[i];
  if (threadIdx.x < C_HID) sB[threadIdx.x] = b1[threadIdx.x];
  __syncthreads();

  const int wave = threadIdx.x >> 5;
  const int lane = threadIdx.x & 31;
  const int rowTile = blockIdx.x * 8 + wave;
  const int row0 = rowTile << 4;
  if (row0 >= M) return;

  const int mn = lane & 15;
  const int kk = (lane >> 4) << 1;
  const float* __restrict__ ar = agg1 + (size_t)(row0 + mn) * C_HID;

  v8f acc = {};
  for (int k = 0; k < C_HID; k += 4) {
    v2f a;
    a.x = fmaxf(ar[k + kk]     + sB[k + kk],     0.0f);
    a.y = fmaxf(ar[k + kk + 1] + sB[k + kk + 1], 0.0f);
    v2f b;
    b.x = sW[(k + kk) * C_OUT + mn];
    b.y = sW[(k + kk + 1) * C_OUT + mn];
    acc = __builtin_amdgcn_wmma_f32_16x16x4_f32(false, a, false, b, (short)0, acc, false, false);
  }

  const int n = lane & 15;
  const int mbase = (lane >> 4) << 3;
  for (int r = 0; r < 8; ++r)
    h2[(size_t)(row0 + mbase + r) * C_OUT + n] = acc[r];
}

// ---------------- layer-2 aggregation (accumulate directly into d_out) ------------

__global__ void k_self2(const float* __restrict__ h2, const float* __restrict__ dinv,
                        float* __restrict__ out, int n) {
  int idx = blockIdx.x * blockDim.x + threadIdx.x;
  if (idx >= n * C_OUT) return;
  int node = idx >> 4;
  float di = dinv[node];
  out[idx] = h2[idx] * di * di;
}

__global__ void k_scatter2(const long long* __restrict__ src,
                           const long long* __restrict__ dst, long long E,
                           const float* __restrict__ dinv,
                           const float* __restrict__ h2,
                           float* __restrict__ out) {
  long long t = (long long)blockIdx.x * blockDim.x + threadIdx.x;
  long long e = t >> 4;
  int f = (int)(t & 15);
  if (e >= E) return;
  if (f == 0 && e + 2048 < E) {
    __builtin_prefetch(&src[e + 2048], 0, 0);
    __builtin_prefetch(&dst[e + 2048], 0, 0);
  }
  int s = (int)src[e], d = (int)dst[e];
  float nrm = dinv[s] * dinv[d];
  atomicAdd(&out[(size_t)d * C_OUT + f], h2[(size_t)s * C_OUT + f] * nrm);
}

__global__ void k_final(float* __restrict__ out, const float* __restrict__ b2, int n) {
  int idx = blockIdx.x * blockDim.x + threadIdx.x;
  if (idx >= n * C_OUT) return;
  out[idx] = fmaxf(out[idx] + b2[idx & 15], 0.0f);
}

// ---------------- launch ----------------

extern "C" void kernel_launch(void* const* d_in, const int* in_sizes, int n_in,
                              void* d_out, int out_size, void* d_ws, size_t ws_size,
                              hipStream_t stream) {
  const float* x        = (const float*)d_in[0];
  const long long* ei   = (const long long*)d_in[1];
  const float* W1       = (const float*)d_in[2];
  const float* b1       = (const float*)d_in[3];
  const float* W2       = (const float*)d_in[4];
  const float* b2       = (const float*)d_in[5];

  const int  N = in_sizes[0] / C_IN;          // 100000
  const long long E = (long long)in_sizes[1] / 2;  // 1600000
  const long long* esrc = ei;
  const long long* edst = ei + E;

  // workspace layout (needs (1+48+48+16)*N*4 ~= 45.2 MB)
  float* dinv = (float*)d_ws;
  float* h1   = dinv + N;
  float* agg1 = h1 + (size_t)N * C_HID;
  float* h2   = agg1 + (size_t)N * C_HID;
  float* out  = (float*)d_out;                // doubles as agg2

  const int BT = 256;
  const int rowTiles = (N + 15) / 16;         // 6250
  const int gemmGrid = (rowTiles + 7) / 8;    // 782

  k_deg_init<<<(N + BT - 1) / BT, BT, 0, stream>>>(dinv, N);
  k_deg_edges<<<(int)((E + BT - 1) / BT), BT, 0, stream>>>(edst, E, dinv);
  k_dinv<<<(N + BT - 1) / BT, BT, 0, stream>>>(dinv, N);

  k_gemm1<<<gemmGrid, BT, 0, stream>>>(x, W1, h1, N);
  k_self1<<<((size_t)N * C_HID + BT - 1) / BT, BT, 0, stream>>>(h1, dinv, agg1, N);
  k_scatter1<<<(int)((E * 16 + BT - 1) / BT), BT, 0, stream>>>(esrc, edst, E, dinv, h1, agg1);

  k_gemm2<<<gemmGrid, BT, 0, stream>>>(agg1, b1, W2, h2, N);
  k_self2<<<((size_t)N * C_OUT + BT - 1) / BT, BT, 0, stream>>>(h2, dinv, out, N);
  k_scatter2<<<(int)((E * 16 + BT - 1) / BT), BT, 0, stream>>>(esrc, edst, E, dinv, h2, out);

  k_final<<<((size_t)N * C_OUT + BT - 1) / BT, BT, 0, stream>>>(out, b2, N);
}